// GNN_60902636257832
// MI455X (gfx1250) — compile-verified
//
#include <hip/hip_runtime.h>
#include <hip/hip_bf16.h>

typedef float v2f __attribute__((ext_vector_type(2)));
typedef float v8f __attribute__((ext_vector_type(8)));

#define NNODES 4096   // B*N
#define KNBR   16
#define ACLD   1024   // A (512) | C (512) per node
#define OUTLD  1920   // 384 + 512 + 512 + 512

// ---------------------------------------------------------------------------
// Geometry MLP: h = relu(relu(rois@gW1.T+gb1)@gW2.T+gb2); X0 = [pooled | h]
// One 128-thread block per node.  Also writes feats[0] into the output slab.
// ---------------------------------------------------------------------------
__global__ void geo_kernel(const float* __restrict__ rois,
                           const float* __restrict__ pooled,
                           const float* __restrict__ gW1, const float* __restrict__ gb1,
                           const float* __restrict__ gW2, const float* __restrict__ gb2,
                           float* __restrict__ X0, float* __restrict__ out)
{
    __shared__ float r[8];
    __shared__ float h1[128];
    const int node = blockIdx.x;
    const int t = threadIdx.x;

    if (t < 7) r[t] = rois[node * 7 + t];
    __syncthreads();

    float s = gb1[t];
    #pragma unroll
    for (int k = 0; k < 7; ++k) s += r[k] * gW1[t * 7 + k];
    h1[t] = fmaxf(s, 0.0f);
    __syncthreads();

    float s2 = gb2[t];
    #pragma unroll 8
    for (int k = 0; k < 128; ++k) s2 += h1[k] * gW2[t * 128 + k];
    const float g = fmaxf(s2, 0.0f);

    // X0 row layout: [pooled(256) | g(128)], stride 384
    X0[node * 384 + 256 + t] = g;
    out[node * OUTLD + 256 + t] = g;

    const float p0 = pooled[node * 256 + t];
    const float p1 = pooled[node * 256 + 128 + t];
    X0[node * 384 + t]        = p0;
    X0[node * 384 + 128 + t]  = p1;
    out[node * OUTLD + t]       = p0;
    out[node * OUTLD + 128 + t] = p1;
}

// ---------------------------------------------------------------------------
// Build Wcomb[1024, dim]: rows 0..511 = Wa, rows 512..1023 = Wb - Wa
// where W = [Wa | Wb] row-major (512, 2*dim).
// ---------------------------------------------------------------------------
__global__ void prepw_kernel(const float* __restrict__ W, int dim,
                             float* __restrict__ Wc)
{
    const int idx = blockIdx.x * blockDim.x + threadIdx.x;
    if (idx >= 512 * dim) return;
    const int o = idx / dim;
    const int k = idx - o * dim;
    const float wa = W[o * (2 * dim) + k];
    const float wb = W[o * (2 * dim) + dim + k];
    Wc[o * dim + k]          = wa;
    Wc[(o + 512) * dim + k]  = wb - wa;
}

// ---------------------------------------------------------------------------
// Node GEMM via V_WMMA_F32_16X16X4_F32 (exact f32 on the matrix pipe):
//   AC[4096, 1024] = X[4096, dim] @ Wc.T      (+ bias folded into cols >= 512)
// One wave owns a 16(M) x 64(N) strip: 4 accumulators, A-fragment reused 4x.
// Fragment layout (ISA 7.12.2, f32 16x4): lane L holds (row=L%16,
// k = 2*(L/16)+{0,1}) as a float2; B mirrors it; C/D: row = v + 8*(L/16),
// col = L%16.
// ---------------------------------------------------------------------------
__global__ void __launch_bounds__(256)
gemm_kernel(const float* __restrict__ X, int ldX, int dim,
            const float* __restrict__ Wc, const float* __restrict__ bias,
            float* __restrict__ AC)
{
    const int wave   = blockIdx.x * (blockDim.x >> 5) + (threadIdx.x >> 5);
    const int lane   = threadIdx.x & 31;
    const int mtile  = wave & 255;      // 256 M-tiles of 16 rows
    const int nstrip = wave >> 8;       // 16 N-strips of 64 cols
    const int m0 = mtile << 4;
    const int n0 = nstrip << 6;
    const int lr = lane & 15;
    const int koff = (lane >> 4) << 1;  // 0 or 2

    const float* xp  = X  + (size_t)(m0 + lr) * ldX + koff;
    const float* wp0 = Wc + (size_t)(n0 +  0 + lr) * dim + koff;
    const float* wp1 = Wc + (size_t)(n0 + 16 + lr) * dim + koff;
    const float* wp2 = Wc + (size_t)(n0 + 32 + lr) * dim + koff;
    const float* wp3 = Wc + (size_t)(n0 + 48 + lr) * dim + koff;

    v8f acc[4] = {};

    #pragma unroll 4
    for (int k = 0; k < dim; k += 4) {
        const v2f a  = *(const v2f*)(xp  + k);
        const v2f b0 = *(const v2f*)(wp0 + k);
        const v2f b1 = *(const v2f*)(wp1 + k);
        const v2f b2 = *(const v2f*)(wp2 + k);
        const v2f b3 = *(const v2f*)(wp3 + k);
        acc[0] = __builtin_amdgcn_wmma_f32_16x16x4_f32(false, a, false, b0,
                                                       (short)0, acc[0], false, false);
        acc[1] = __builtin_amdgcn_wmma_f32_16x16x4_f32(false, a, false, b1,
                                                       (short)0, acc[1], false, false);
        acc[2] = __builtin_amdgcn_wmma_f32_16x16x4_f32(false, a, false, b2,
                                                       (short)0, acc[2], false, false);
        acc[3] = __builtin_amdgcn_wmma_f32_16x16x4_f32(false, a, false, b3,
                                                       (short)0, acc[3], false, false);
    }

    const int rbase = m0 + ((lane >> 4) << 3);
    #pragma unroll
    for (int t = 0; t < 4; ++t) {
        const int col = n0 + t * 16 + lr;
        const float badd = (col >= 512) ? bias[col - 512] : 0.0f;
        #pragma unroll
        for (int v = 0; v < 8; ++v) {
            AC[(size_t)(rbase + v) * ACLD + col] = acc[t][v] + badd;
        }
    }
}

// ---------------------------------------------------------------------------
// Per-node aggregation:  x_new[i] = relu( max_j A[src_j]  +  C[i] )
// (relu/max commute since C[i] is constant over j; bias already in C).
// Edges for node i are contiguous: e = i*16 + k, neighbor = src[e].
// ---------------------------------------------------------------------------
__global__ void agg_kernel(const float* __restrict__ AC,
                           const int* __restrict__ src,
                           float* __restrict__ Xn,
                           float* __restrict__ out, int colOff)
{
    __shared__ int nb[KNBR];
    const int node = blockIdx.x;
    const int t = threadIdx.x;
    if (t < KNBR) nb[t] = src[node * KNBR + t];
    __syncthreads();

    for (int o = t; o < 512; o += blockDim.x) {
        float m = -3.402823466e38f;
        #pragma unroll
        for (int k = 0; k < KNBR; ++k)
            m = fmaxf(m, AC[(size_t)nb[k] * ACLD + o]);
        const float v = fmaxf(m + AC[(size_t)node * ACLD + 512 + o], 0.0f);
        Xn[node * 512 + o] = v;
        out[node * OUTLD + colOff + o] = v;
    }
}

// ---------------------------------------------------------------------------
extern "C" void kernel_launch(void* const* d_in, const int* in_sizes, int n_in,
                              void* d_out, int out_size, void* d_ws, size_t ws_size,
                              hipStream_t stream)
{
    const float* rois   = (const float*)d_in[0];
    const float* pooled = (const float*)d_in[1];
    const int*   edge   = (const int*)d_in[2];     // row 0 = src (65536 ints)
    const float* gW1 = (const float*)d_in[3];
    const float* gb1 = (const float*)d_in[4];
    const float* gW2 = (const float*)d_in[5];
    const float* gb2 = (const float*)d_in[6];
    const float* fcW[3] = { (const float*)d_in[7], (const float*)d_in[9],  (const float*)d_in[11] };
    const float* fcb[3] = { (const float*)d_in[8], (const float*)d_in[10], (const float*)d_in[12] };
    float* out = (float*)d_out;

    // Workspace: X (4096x512 f32, 8MB) | AC (4096x1024 f32, 16MB) | Wc (1024x512, 2MB)
    float* Xbuf = (float*)d_ws;
    float* AC   = Xbuf + (size_t)NNODES * 512;
    float* Wc   = AC   + (size_t)NNODES * 1024;

    geo_kernel<<<NNODES, 128, 0, stream>>>(rois, pooled, gW1, gb1, gW2, gb2, Xbuf, out);

    const int dims[3]   = { 384, 512, 512 };
    const int colOff[3] = { 384, 896, 1408 };
    for (int l = 0; l < 3; ++l) {
        const int dim = dims[l];
        const int nprep = 512 * dim;
        prepw_kernel<<<(nprep + 255) / 256, 256, 0, stream>>>(fcW[l], dim, Wc);
        // 4096 waves total: 256 M-tiles x 16 N-strips, 8 waves per block
        gemm_kernel<<<512, 256, 0, stream>>>(Xbuf, (l == 0) ? 384 : 512, dim, Wc, fcb[l], AC);
        agg_kernel<<<NNODES, 256, 0, stream>>>(AC, edge, Xbuf, out, colOff[l]);
    }
}